// SetAbstractionLayer_89979564851532
// MI455X (gfx1250) — compile-verified
//
#include <hip/hip_runtime.h>
#include <hip/hip_bf16.h>

typedef float v2f __attribute__((ext_vector_type(2)));
typedef float v8f __attribute__((ext_vector_type(8)));

#define NPTS          65536
#define NBATCH        16
#define NSAMPLE       2048
#define WGS_PER_BATCH 16
#define THREADS       256
#define TILES         32   // per wave: 32 tiles * 16 pts = 512 pts; 8 waves -> 4096 pts per WG

__global__ void fps_init_kernel(int* __restrict__ cnt) {
  if (threadIdx.x < NBATCH) cnt[threadIdx.x] = 0;
}

__global__ __launch_bounds__(THREADS)
void fps_kernel(const float* __restrict__ coords, float* __restrict__ out,
                int* __restrict__ cnt, float* __restrict__ pval, int* __restrict__ pidx) {
  const int b    = blockIdx.x >> 4;        // batch
  const int wgi  = blockIdx.x & 15;        // workgroup within batch
  const int tid  = threadIdx.x;
  const int lane = tid & 31;
  const int wave = tid >> 5;
  const bool lower = (lane < 16);
  const int lm   = lane & 15;
  const float* pc = coords + (size_t)b * NPTS * 3;

  // ---- load this wave's 512 points once, directly in WMMA B-layout ----
  // B (4x16 f32, 2 VGPRs): VGPR0 = {K0 | K2} on lane halves, VGPR1 = {K1 | K3}.
  // Rows: K0=x, K1=y, K2=z, K3=1  ->  lanes<16: (x,y); lanes>=16: (z,1).
  v2f   bv[TILES];
  float pn[TILES];   // |p|^2 for point tbase+lm (duplicated on upper half-wave)
  float cl[TILES];   // running min distance, same lane mapping
  const int wavebase = wgi * 4096 + wave * 512;
#pragma unroll
  for (int t = 0; t < TILES; ++t) {
    const int j = wavebase + t * 16 + lm;
    const float x = pc[j * 3 + 0];
    const float y = pc[j * 3 + 1];
    const float z = pc[j * 3 + 2];
    pn[t]   = x * x + y * y + z * z;
    bv[t].x = lower ? x : z;
    bv[t].y = lower ? y : 1.0f;
    cl[t]   = __builtin_inff();
  }

  __shared__ float s_sel[3];
  __shared__ float s_wv[THREADS / 32];
  __shared__ int   s_wi[THREADS / 32];
  if (tid == 0) { s_sel[0] = pc[0]; s_sel[1] = pc[1]; s_sel[2] = pc[2]; }
  __syncthreads();
  float sx = s_sel[0], sy = s_sel[1], sz = s_sel[2];

  for (int it = 0; it < NSAMPLE; ++it) {
    // A (16x4 f32, all rows identical): (-2sx, -2sy, -2sz, |s|^2)
    // Layout: lanes<16: {K0,K1} = (-2sx,-2sy); lanes>=16: {K2,K3} = (-2sz,|s|^2)
    const float snorm = sx * sx + sy * sy + sz * sz;
    v2f a;
    a.x = lower ? (-2.0f * sx) : (-2.0f * sz);
    a.y = lower ? (-2.0f * sy) : snorm;

    float bestv = -__builtin_inff();
    int   bestt = 0;
#pragma unroll
    for (int t = 0; t < TILES; ++t) {
      v8f c = {};
      // D[m][n] = |s|^2 - 2 s.p_n  (rows identical); lane L reads result for
      // point tbase + (L&15) from element 0 on both lane halves.
      v8f d = __builtin_amdgcn_wmma_f32_16x16x4_f32(
          /*neg_a=*/false, a, /*neg_b=*/false, bv[t],
          /*c_mod=*/(short)0, c, /*reuse_a=*/false, /*reuse_b=*/false);
      const float dist = d[0] + pn[t];                 // full squared distance
      // no-NaN data: plain compare+select avoids v_max canonicalize traffic
      const float m = (dist < cl[t]) ? dist : cl[t];
      cl[t] = m;
      if (m > bestv) { bestv = m; bestt = t; }         // strict > keeps smallest index on ties
    }
    int besti = wavebase + bestt * 16 + lm;

    // ---- wave argmax (tie -> smaller index; upper half-wave is a duplicate) ----
#pragma unroll
    for (int m = 16; m >= 1; m >>= 1) {
      const float ov = __shfl_xor(bestv, m, 32);
      const int   oi = __shfl_xor(besti, m, 32);
      if (ov > bestv || (ov == bestv && oi < besti)) { bestv = ov; besti = oi; }
    }
    if (lane == 0) { s_wv[wave] = bestv; s_wi[wave] = besti; }
    __syncthreads();

    if (tid == 0) {
      // ---- inter-wave argmax ----
      float wv = s_wv[0]; int wi = s_wi[0];
#pragma unroll
      for (int k = 1; k < THREADS / 32; ++k)
        if (s_wv[k] > wv || (s_wv[k] == wv && s_wi[k] < wi)) { wv = s_wv[k]; wi = s_wi[k]; }

      // ---- publish partial, arrive at monotonic batch barrier ----
      const int p    = it & 1;                              // double-buffered partials
      const int slot = p * (NBATCH * WGS_PER_BATCH) + b * WGS_PER_BATCH + wgi;
      __hip_atomic_store(&pval[slot], wv, __ATOMIC_RELAXED, __HIP_MEMORY_SCOPE_AGENT);
      __hip_atomic_store(&pidx[slot], wi, __ATOMIC_RELAXED, __HIP_MEMORY_SCOPE_AGENT);
      __hip_atomic_fetch_add(&cnt[b], 1, __ATOMIC_ACQ_REL, __HIP_MEMORY_SCOPE_AGENT);
      const int target = (it + 1) * WGS_PER_BATCH;          // monotonic: no reset races
      while (__hip_atomic_load(&cnt[b], __ATOMIC_ACQUIRE, __HIP_MEMORY_SCOPE_AGENT) < target)
        __builtin_amdgcn_s_sleep(8);

      // ---- batch-wide winner (every WG computes it identically) ----
      const int base = p * (NBATCH * WGS_PER_BATCH) + b * WGS_PER_BATCH;
      float gv = __hip_atomic_load(&pval[base], __ATOMIC_RELAXED, __HIP_MEMORY_SCOPE_AGENT);
      int   gi = __hip_atomic_load(&pidx[base], __ATOMIC_RELAXED, __HIP_MEMORY_SCOPE_AGENT);
      for (int k = 1; k < WGS_PER_BATCH; ++k) {
        const float v = __hip_atomic_load(&pval[base + k], __ATOMIC_RELAXED, __HIP_MEMORY_SCOPE_AGENT);
        const int   i = __hip_atomic_load(&pidx[base + k], __ATOMIC_RELAXED, __HIP_MEMORY_SCOPE_AGENT);
        if (v > gv || (v == gv && i < gi)) { gv = v; gi = i; }
      }
      const float nx = pc[(size_t)gi * 3 + 0];
      const float ny = pc[(size_t)gi * 3 + 1];
      const float nz = pc[(size_t)gi * 3 + 2];
      s_sel[0] = nx; s_sel[1] = ny; s_sel[2] = nz;
      if (wgi == 0) {
        const size_t o = ((size_t)b * NSAMPLE + it) * 3;
        out[o + 0] = nx; out[o + 1] = ny; out[o + 2] = nz;
      }
    }
    __syncthreads();
    sx = s_sel[0]; sy = s_sel[1]; sz = s_sel[2];
  }
}

extern "C" void kernel_launch(void* const* d_in, const int* in_sizes, int n_in,
                              void* d_out, int out_size, void* d_ws, size_t ws_size,
                              hipStream_t stream) {
  const float* coords = (const float*)d_in[0];   // (16, 65536, 3) f32
  // d_in[1] = features, unused by the reference output path
  float* out = (float*)d_out;                    // (16, 2048, 3) f32

  // workspace layout (tiny): [0,64B) counters | [256B) pval[2][256] | then pidx[2][256]
  int*   cnt  = (int*)d_ws;
  float* pval = (float*)((char*)d_ws + 256);
  int*   pidx = (int*)((char*)d_ws + 256 + 2 * NBATCH * WGS_PER_BATCH * sizeof(float));

  fps_init_kernel<<<1, 64, 0, stream>>>(cnt);
  fps_kernel<<<NBATCH * WGS_PER_BATCH, THREADS, 0, stream>>>(coords, out, cnt, pval, pidx);
}